// GATLayer_188978561180
// MI455X (gfx1250) — compile-verified
//
#include <hip/hip_runtime.h>
#include <stdint.h>

#define NUM_NODES 100000
#define NUM_EDGES 1000000
#define IN_FEAT   128
#define NUM_HEADS 4
#define OUT_FEAT  16
#define OUT_TOT   64          // NUM_HEADS * OUT_FEAT
#define ALPHA     0.2f
#define M_TILES   (NUM_NODES / 16)   // 6250, exact

typedef __attribute__((ext_vector_type(2))) float v2f;
typedef __attribute__((ext_vector_type(8))) float v8f;

// ---------------------------------------------------------------------------
// Kernel 1: Wh = h @ W   via V_WMMA_F32_16X16X4_F32
// One wave -> 16 rows x 64 cols (4 N-tiles of 16). 8 waves / block.
// ---------------------------------------------------------------------------
__global__ __launch_bounds__(256) void gat_gemm_wmma(
    const float* __restrict__ h, const float* __restrict__ W,
    float* __restrict__ Wh) {
  const int lane  = threadIdx.x & 31;
  const int wave  = threadIdx.x >> 5;
  const int tileM = blockIdx.x * 8 + wave;
  if (tileM >= M_TILES) return;               // wave-uniform: EXEC stays all-1

  const int half = lane >> 4;                 // 0: lanes 0-15, 1: lanes 16-31
  const int fcol = lane & 15;
  const float* __restrict__ hrow = h + (size_t)(tileM * 16 + fcol) * IN_FEAT;

  v8f c0 = {}, c1 = {}, c2 = {}, c3 = {};

  #pragma unroll 8
  for (int k = 0; k < IN_FEAT; k += 4) {
    const int kk = k + 2 * half;              // this lane's K pair: kk, kk+1
    // A fragment: A[M=fcol][K=kk], A[M=fcol][K=kk+1]
    const v2f a = *(const v2f*)(hrow + kk);
    // B fragments: B[K=kk][N], B[K=kk+1][N] for each 16-col head tile
    const float* __restrict__ w0 = W + (size_t)kk * OUT_TOT + fcol;
    const float* __restrict__ w1 = w0 + OUT_TOT;
    v2f b0, b1, b2, b3;
    b0.x = w0[0];  b0.y = w1[0];
    b1.x = w0[16]; b1.y = w1[16];
    b2.x = w0[32]; b2.y = w1[32];
    b3.x = w0[48]; b3.y = w1[48];
    c0 = __builtin_amdgcn_wmma_f32_16x16x4_f32(false, a, false, b0, (short)0, c0, false, false);
    c1 = __builtin_amdgcn_wmma_f32_16x16x4_f32(false, a, false, b1, (short)0, c1, false, false);
    c2 = __builtin_amdgcn_wmma_f32_16x16x4_f32(false, a, false, b2, (short)0, c2, false, false);
    c3 = __builtin_amdgcn_wmma_f32_16x16x4_f32(false, a, false, b3, (short)0, c3, false, false);
  }

  // D layout: VGPR r -> M = r (lanes 0-15) / M = r+8 (lanes 16-31), N = fcol
  const int rowBase = tileM * 16 + half * 8;
  #pragma unroll
  for (int r = 0; r < 8; ++r) {
    float* __restrict__ o = Wh + (size_t)(rowBase + r) * OUT_TOT + fcol;
    o[0]  = c0[r];
    o[16] = c1[r];
    o[32] = c2[r];
    o[48] = c3[r];
  }
}

// ---------------------------------------------------------------------------
// Kernel 2: per-node attention logit halves
//   s_src[n,h] = sum_f Wh[n,h,f]*a_src[h,f] ; same for s_dst
// ---------------------------------------------------------------------------
__global__ __launch_bounds__(256) void gat_node_scores(
    const float* __restrict__ Wh, const float* __restrict__ a_src,
    const float* __restrict__ a_dst, float* __restrict__ s_src,
    float* __restrict__ s_dst) {
  const int n = blockIdx.x * 256 + threadIdx.x;
  if (n >= NUM_NODES) return;
  const float* __restrict__ row = Wh + (size_t)n * OUT_TOT;
  float sa[NUM_HEADS], sb[NUM_HEADS];
  #pragma unroll
  for (int hh = 0; hh < NUM_HEADS; ++hh) {
    float acc1 = 0.f, acc2 = 0.f;
    #pragma unroll
    for (int f = 0; f < OUT_FEAT; ++f) {
      const float w = row[hh * OUT_FEAT + f];
      acc1 += w * a_src[hh * OUT_FEAT + f];
      acc2 += w * a_dst[hh * OUT_FEAT + f];
    }
    sa[hh] = acc1; sb[hh] = acc2;
  }
  ((float4*)s_src)[n] = make_float4(sa[0], sa[1], sa[2], sa[3]);
  ((float4*)s_dst)[n] = make_float4(sb[0], sb[1], sb[2], sb[3]);
}

// ---------------------------------------------------------------------------
// Hardware fp32 scatter-add: device-scope L2 RMW, non-returning (TH[0]=0).
// Inline asm guarantees global_atomic_add_f32 (no CAS-loop expansion), and
// SCOPE_DEV guarantees the RMW happens at a device-coherent level (a default
// WGP-scope atomic could legally RMW in the WGP cache -> lost updates).
// ---------------------------------------------------------------------------
__device__ __forceinline__ void atomicAddDevF32(float* p, float v) {
  asm volatile("global_atomic_add_f32 %0, %1, off scope:SCOPE_DEV"
               :: "v"(p), "v"(v)
               : "memory");
}

// ---------------------------------------------------------------------------
// Kernel 3: per-edge softmax-over-heads + scatter-add.
// One edge per half-wave: lane f = 0..15 covers feature f of all 4 heads,
// so each head's 16 atomics land on one contiguous 64B span.
// ---------------------------------------------------------------------------
__global__ __launch_bounds__(256) void gat_edge_scatter(
    const int64_t* __restrict__ ei, const float* __restrict__ Wh,
    const float* __restrict__ s_src, const float* __restrict__ s_dst,
    float* __restrict__ out) {
  const int lane = threadIdx.x & 31;
  const int wave = threadIdx.x >> 5;
  const int half = lane >> 4;
  const int f    = lane & 15;
  const long long e = (long long)blockIdx.x * 16 + wave * 2 + half;
  if (e >= NUM_EDGES) return;

  const int src = (int)ei[e];
  const int dst = (int)ei[NUM_EDGES + e];

  const float4 ss = ((const float4*)s_src)[src];
  const float4 sd = ((const float4*)s_dst)[dst];
  float eh[NUM_HEADS] = {ss.x + sd.x, ss.y + sd.y, ss.z + sd.z, ss.w + sd.w};
  #pragma unroll
  for (int hh = 0; hh < NUM_HEADS; ++hh)
    eh[hh] = eh[hh] >= 0.f ? eh[hh] : ALPHA * eh[hh];
  // softmax over the HEADS axis (per edge) -- faithful to reference
  const float m = fmaxf(fmaxf(eh[0], eh[1]), fmaxf(eh[2], eh[3]));
  float s = 0.f;
  #pragma unroll
  for (int hh = 0; hh < NUM_HEADS; ++hh) { eh[hh] = __expf(eh[hh] - m); s += eh[hh]; }
  const float inv = 1.f / s;

  const float* __restrict__ wsrc = Wh + (size_t)src * OUT_TOT;
  float* __restrict__ od = out + (size_t)dst * OUT_TOT;
  #pragma unroll
  for (int hh = 0; hh < NUM_HEADS; ++hh) {
    const float v = eh[hh] * inv * wsrc[hh * OUT_FEAT + f];
    atomicAddDevF32(od + hh * OUT_FEAT + f, v);
  }
}

// ---------------------------------------------------------------------------
// Kernel 0/4: zero the accumulator, final ReLU (in place)
// ---------------------------------------------------------------------------
__global__ __launch_bounds__(256) void gat_zero(float4* __restrict__ p, int n4) {
  const int i = blockIdx.x * 256 + threadIdx.x;
  if (i < n4) p[i] = make_float4(0.f, 0.f, 0.f, 0.f);
}

__global__ __launch_bounds__(256) void gat_relu(float4* __restrict__ p, int n4) {
  const int i = blockIdx.x * 256 + threadIdx.x;
  if (i < n4) {
    float4 v = p[i];
    v.x = fmaxf(v.x, 0.f); v.y = fmaxf(v.y, 0.f);
    v.z = fmaxf(v.z, 0.f); v.w = fmaxf(v.w, 0.f);
    p[i] = v;
  }
}

// ---------------------------------------------------------------------------
extern "C" void kernel_launch(void* const* d_in, const int* in_sizes, int n_in,
                              void* d_out, int out_size, void* d_ws, size_t ws_size,
                              hipStream_t stream) {
  const float*   h     = (const float*)d_in[0];
  const int64_t* ei    = (const int64_t*)d_in[1];   // [2, E] int64 per reference
  const float*   W     = (const float*)d_in[2];     // [128, 64] row-major
  const float*   a_src = (const float*)d_in[3];     // [4, 16]
  const float*   a_dst = (const float*)d_in[4];
  float* out = (float*)d_out;

  // workspace layout: Wh (25.6MB) | s_src (1.6MB) | s_dst (1.6MB)
  float* Wh    = (float*)d_ws;
  float* s_src = Wh + (size_t)NUM_NODES * OUT_TOT;
  float* s_dst = s_src + (size_t)NUM_NODES * NUM_HEADS;

  const int n4 = NUM_NODES * OUT_TOT / 4;                       // 1.6M float4
  gat_zero<<<(n4 + 255) / 256, 256, 0, stream>>>((float4*)out, n4);

  gat_gemm_wmma<<<(M_TILES + 7) / 8, 256, 0, stream>>>(h, W, Wh);

  gat_node_scores<<<(NUM_NODES + 255) / 256, 256, 0, stream>>>(Wh, a_src, a_dst,
                                                               s_src, s_dst);

  gat_edge_scatter<<<(NUM_EDGES + 15) / 16, 256, 0, stream>>>(ei, Wh, s_src,
                                                              s_dst, out);

  gat_relu<<<(n4 + 255) / 256, 256, 0, stream>>>((float4*)out, n4);
}